// SeparationLoss_31602369364353
// MI455X (gfx1250) — compile-verified
//
#include <hip/hip_runtime.h>

// D = A(16x4) x B(4x16) + C, fp32 WMMA on gfx1250 (wave32).
typedef float v2f __attribute__((ext_vector_type(2)));
typedef float v8f __attribute__((ext_vector_type(8)));

// exp(-d/25) = exp2(d * (-log2(e)/25))
#define NEG_L2E_OVER_S2 (-0.05770780163555854f)

__global__ void sep_loss_zero_kernel(float* out) {
    out[0] = 0.0f;
}

// Grid: (17, B). Block: 256 threads = 8 waves. Each wave computes one
// 16x16 upper-triangular tile of the NxN pairwise-distance matrix via a
// single V_WMMA_F32_16X16X4_F32, applies d2*exp(-d2/25) with the i<j mask,
// and reduces.
__global__ __launch_bounds__(256)
void sep_loss_kernel(const float* __restrict__ landmarks, float* __restrict__ out) {
    constexpr int N = 256;

    const int b    = blockIdx.y;
    const int wave = threadIdx.x >> 5;   // 0..7
    const int lane = threadIdx.x & 31;   // wave32
    const int half = lane >> 4;          // 0: K=0/1 half, 1: K=2/3 half
    const int l    = lane & 15;

    // Upper-triangular tile index t in [0,136); row I holds (16-I) tiles.
    int t = blockIdx.x * 8 + wave;
    int I = 0, rem = 16;
    while (t >= rem) { t -= rem; ++I; --rem; }
    const int J = I + t;                 // I <= J

    // Points for this batch: [N] float2, contiguous.
    const float2* pts = reinterpret_cast<const float2*>(landmarks) + (size_t)b * N;
    const float2 pi = pts[I * 16 + l];   // row block
    const float2 pj = pts[J * 16 + l];   // col block
    const float si = pi.x * pi.x + pi.y * pi.y;
    const float sj = pj.x * pj.x + pj.y * pj.y;

    // A row i  = [x_i, y_i, s_i, 1];  B col j = [-2x_j, -2y_j, 1, s_j]
    // => (A*B)[i][j] = s_i + s_j - 2*dot(p_i,p_j) = |p_i - p_j|^2 exactly.
    // A layout (16x4 f32): lanes 0-15 -> {K0,K1}, lanes 16-31 -> {K2,K3}.
    // Branch-free selects keep EXEC all-1s for the WMMA.
    v2f a, bb;
    a.x  = half ? si            : pi.x;
    a.y  = half ? 1.0f          : pi.y;
    bb.x = half ? 1.0f          : -2.0f * pj.x;
    bb.y = half ? sj            : -2.0f * pj.y;

    v8f c = {};
    c = __builtin_amdgcn_wmma_f32_16x16x4_f32(
        /*neg_a=*/false, a, /*neg_b=*/false, bb,
        /*c_mod=*/(short)0, c, /*reuse_a=*/false, /*reuse_b=*/false);

    // D tile layout: VGPR r, lane -> (M = r + 8*half, N = l).
    float acc = 0.0f;
#pragma unroll
    for (int r = 0; r < 8; ++r) {
        const int gi = I * 16 + r + 8 * half;
        const int gj = J * 16 + l;
        float d = fmaxf(c[r], 0.0f);                       // numerical floor
        float w = (gi < gj) ? d : 0.0f;                    // strict upper triangle
        acc += w * __builtin_amdgcn_exp2f(d * NEG_L2E_OVER_S2);
    }

    // Wave32 reduction, then block reduction, then one atomic per block.
#pragma unroll
    for (int off = 16; off > 0; off >>= 1)
        acc += __shfl_down(acc, off, 32);

    __shared__ float red[8];
    if (lane == 0) red[wave] = acc;
    __syncthreads();
    if (threadIdx.x == 0) {
        float s = 0.0f;
#pragma unroll
        for (int w = 0; w < 8; ++w) s += red[w];
        atomicAdd(out, s);   // LAMDA == 1.0
    }
}

extern "C" void kernel_launch(void* const* d_in, const int* in_sizes, int n_in,
                              void* d_out, int out_size, void* d_ws, size_t ws_size,
                              hipStream_t stream) {
    (void)in_sizes; (void)n_in; (void)out_size; (void)d_ws; (void)ws_size;
    const float* landmarks = (const float*)d_in[0];
    float* out = (float*)d_out;

    // d_out is poisoned by the harness; zero it first (same stream => ordered).
    sep_loss_zero_kernel<<<1, 1, 0, stream>>>(out);

    // 1024 batches x 136 upper-tri tiles = grid(17, 1024), 8 tiles (waves) per block.
    dim3 grid(17, 1024);
    sep_loss_kernel<<<grid, 256, 0, stream>>>(landmarks, out);
}